// PO2Vec_30382598651986
// MI455X (gfx1250) — compile-verified
//
#include <hip/hip_runtime.h>

typedef __attribute__((ext_vector_type(2))) float v2f;
typedef __attribute__((ext_vector_type(8))) float v8f;

namespace {
constexpr int   kEmbDim  = 256;
constexpr int   kB       = 1024;
constexpr int   kP       = 64;
constexpr int   kN       = 512;
constexpr int   kCand    = kP + kN;      // 576
constexpr int   kTiles   = kCand / 16;   // 36 tiles of 16 candidates
constexpr float kInvTemp = 10.0f;        // 1 / 0.1
constexpr float kEps     = 1e-8f;
}

__global__ __launch_bounds__(256) void po2vec_main(
    const int*   __restrict__ anchor_idx,
    const int*   __restrict__ pos_idx,
    const int*   __restrict__ neg_idx,
    const float* __restrict__ emb,
    float*       __restrict__ partial)      // [kB] per-block sums
{
    __shared__ __align__(16) float a_lds[kEmbDim];   // anchor row
    __shared__ float s_dot[kCand];                   // X . a
    __shared__ float s_nrm2[kCand];                  // ||x||^2
    __shared__ float s_part[8];

    const int  b    = blockIdx.x;
    const int  tid  = threadIdx.x;
    const int  wave = tid >> 5;
    const int  lane = tid & 31;
    const int  m    = lane & 15;     // candidate row within tile
    const bool hi   = lane >= 16;    // which K half this lane feeds

    // ---------- Phase 1: anchor row -> LDS, ||a||^2 (deterministic tree) ----
    {
        const float av = emb[(size_t)anchor_idx[b] * kEmbDim + tid];
        a_lds[tid] = av;
        float sq = av * av;
        #pragma unroll
        for (int o = 16; o > 0; o >>= 1) sq += __shfl_xor(sq, o, 32);
        if (lane == 0) s_part[wave] = sq;
    }
    __syncthreads();
    float n2a = 0.0f;
    #pragma unroll
    for (int w = 0; w < 8; ++w) n2a += s_part[w];
    const float norm_a = sqrtf(n2a);

    const int* __restrict__ pidx = pos_idx + (size_t)b * kP;
    const int* __restrict__ nidx = neg_idx + (size_t)b * kN;

    // ---------- Phase 2: 16-candidate tiles via V_WMMA_F32_16X16X4_F32 ------
    for (int tile = wave; tile < kTiles; tile += 8) {
        const int cand = tile * 16 + m;
        const int idx  = (cand < kP) ? pidx[cand] : nidx[cand - kP];
        const float* __restrict__ row = emb + (size_t)idx * kEmbDim;
        const int koff = hi ? 2 : 0;

        v8f c0 = {};
        v8f c1 = {};
        float ss = 0.0f;
        #pragma unroll 4
        for (int k0 = 0; k0 < kEmbDim; k0 += 8) {
            {   // even K-step -> accumulator c0
                const int ka = k0 + koff;
                const float2 xv = *(const float2*)(row + ka);     // A frag (gather)
                const float2 bv = *(const float2*)(a_lds + ka);   // B frag (broadcast a)
                ss = fmaf(xv.x, xv.x, ss);
                ss = fmaf(xv.y, xv.y, ss);
                v2f A;  A.x  = xv.x; A.y  = xv.y;
                v2f Bv; Bv.x = bv.x; Bv.y = bv.y;
                c0 = __builtin_amdgcn_wmma_f32_16x16x4_f32(
                        false, A, false, Bv, (short)0, c0, false, false);
            }
            {   // odd K-step -> accumulator c1 (breaks WMMA RAW chain)
                const int ka = k0 + 4 + koff;
                const float2 xv = *(const float2*)(row + ka);
                const float2 bv = *(const float2*)(a_lds + ka);
                ss = fmaf(xv.x, xv.x, ss);
                ss = fmaf(xv.y, xv.y, ss);
                v2f A;  A.x  = xv.x; A.y  = xv.y;
                v2f Bv; Bv.x = bv.x; Bv.y = bv.y;
                c1 = __builtin_amdgcn_wmma_f32_16x16x4_f32(
                        false, A, false, Bv, (short)0, c1, false, false);
            }
        }
        const v8f c = c0 + c1;

        // lane l and l+16 each hold half of ||x_m||^2 for row m = l&15
        ss += __shfl_xor(ss, 16, 32);
        if (lane < 16) s_nrm2[cand] = ss;

        // C layout: VGPR j = dot[j] on lanes 0-15, dot[8+j] on lanes 16-31
        // (all 16 columns identical since B is broadcast)
        if (lane == 0) {
            #pragma unroll
            for (int j = 0; j < 8; ++j) s_dot[tile * 16 + j] = c[j];
        }
        if (lane == 16) {
            #pragma unroll
            for (int j = 0; j < 8; ++j) s_dot[tile * 16 + 8 + j] = c[j];
        }
    }
    __syncthreads();

    // ---------- Phase 3: BCE-with-logits over 576 candidates ----------------
    float acc = 0.0f;
    for (int i = tid; i < kCand; i += 256) {
        const float nx  = sqrtf(s_nrm2[i]);
        const float den = fmaxf(norm_a * nx, kEps);
        const float z   = (s_dot[i] / den) * kInvTemp;
        const float y   = (i < kP) ? 1.0f : 0.0f;
        acc += fmaxf(z, 0.0f) - z * y + log1pf(expf(-fabsf(z)));
    }
    #pragma unroll
    for (int o = 16; o > 0; o >>= 1) acc += __shfl_xor(acc, o, 32);
    if (lane == 0) s_part[wave] = acc;
    __syncthreads();
    if (tid == 0) {
        float t = 0.0f;
        #pragma unroll
        for (int w = 0; w < 8; ++w) t += s_part[w];
        partial[b] = t;
    }
}

// Fixed-order reduction of the 1024 per-block partials -> deterministic scalar.
__global__ __launch_bounds__(256) void po2vec_reduce(
    const float* __restrict__ partial, float* __restrict__ out)
{
    __shared__ float s_part[8];
    const int tid = threadIdx.x;
    float v = partial[tid] + partial[tid + 256] + partial[tid + 512] + partial[tid + 768];
    #pragma unroll
    for (int o = 16; o > 0; o >>= 1) v += __shfl_xor(v, o, 32);
    if ((tid & 31) == 0) s_part[tid >> 5] = v;
    __syncthreads();
    if (tid == 0) {
        float t = 0.0f;
        #pragma unroll
        for (int w = 0; w < 8; ++w) t += s_part[w];
        out[0] = t * (1.0f / ((float)kB * (float)kCand));
    }
}

extern "C" void kernel_launch(void* const* d_in, const int* in_sizes, int n_in,
                              void* d_out, int out_size, void* d_ws, size_t ws_size,
                              hipStream_t stream) {
    (void)in_sizes; (void)n_in; (void)out_size; (void)ws_size;
    const int*   anchor_idx = (const int*)d_in[0];
    const int*   pos_idx    = (const int*)d_in[1];
    const int*   neg_idx    = (const int*)d_in[2];
    const float* emb        = (const float*)d_in[3];
    float* partial = (float*)d_ws;     // kB floats of scratch
    float* out     = (float*)d_out;

    po2vec_main<<<kB, 256, 0, stream>>>(anchor_idx, pos_idx, neg_idx, emb, partial);
    po2vec_reduce<<<1, 256, 0, stream>>>(partial, out);
}